// DGLR_91079076479123
// MI455X (gfx1250) — compile-verified
//
#include <hip/hip_runtime.h>
#include <hip/hip_bf16.h>
#include <math.h>
#include <stdint.h>

// ---------------------------------------------------------------------------
// EvolveGCN-style GRCU: 2 layers of (GAT -> per-node GRU) over T timesteps.
// fp32 matrix work on V_WMMA_F32_16X16X4_F32; attention tiles streamed with
// GLOBAL_LOAD_ASYNC_TO_LDS_B128 double buffering (single-wave workgroups, so
// only s_wait_asynccnt is needed, no barriers).  CDNA5 / gfx1250.
// ---------------------------------------------------------------------------

#define NN      2000
#define TT      8
#define HH      4
#define ALPHA_L 0.2f

typedef float v2f __attribute__((ext_vector_type(2)));
typedef float v8f __attribute__((ext_vector_type(8)));

// ---- CDNA5 async copy helpers (inline asm; builtins differ across toolchains)
__device__ __forceinline__ void async_b128(unsigned lds_off, const float* g) {
    // per-lane: copies 16 bytes from global address g to LDS address lds_off, tracked with ASYNCcnt
    asm volatile("global_load_async_to_lds_b128 %0, %1, off"
                 :: "v"(lds_off), "v"(g) : "memory");
}
template<int N>
__device__ __forceinline__ void wait_async() {
    asm volatile("s_wait_asynccnt %0" :: "i"(N) : "memory");
}

// ---------------------------------------------------------------------------
// Kernel 1: per-head feature projection  h[t,head,:,:] = x[t] @ W[head]
//   x: [T,N,F], W: [H,F,D], Hout: [T,H,N,D].  One wave per (rowblock,head,t).
// ---------------------------------------------------------------------------
template<int F, int D>
__global__ __launch_bounds__(32)
void hproj_kernel(const float* __restrict__ X, const float* __restrict__ W,
                  float* __restrict__ Hout) {
    const int rb = blockIdx.x, head = blockIdx.y, t = blockIdx.z;
    const int lane = threadIdx.x;
    const int m  = lane & 15;
    const int hi = lane >> 4;          // 0: lanes 0-15, 1: lanes 16-31

    __shared__ float xs[16 * F];
    const float* x = X + (size_t)t * NN * F + (size_t)rb * 16 * F;
    for (int i = lane; i < 16 * F / 4; i += 32)
        ((float4*)xs)[i] = ((const float4*)x)[i];
    __syncthreads();

    const float* Wh = W + (size_t)head * F * D;
    float* hout = Hout + (((size_t)t * HH + head) * NN + (size_t)rb * 16) * D;

#pragma unroll
    for (int dt = 0; dt < D / 16; ++dt) {
        v8f acc = {};
#pragma unroll
        for (int kb = 0; kb < F / 4; ++kb) {
            const int k0 = kb * 4 + hi * 2;
            v2f a, b;
            a.x = xs[m * F + k0];                      // A: row m, K=k0
            a.y = xs[m * F + k0 + 1];
            b.x = Wh[(size_t)k0 * D + dt * 16 + m];    // B: K rows, col dt*16+m
            b.y = Wh[(size_t)(k0 + 1) * D + dt * 16 + m];
            acc = __builtin_amdgcn_wmma_f32_16x16x4_f32(
                false, a, false, b, (short)0, acc, false, false);
        }
#pragma unroll
        for (int i = 0; i < 8; ++i)                    // C: vgpr i -> row i+8*hi
            hout[(size_t)(i + hi * 8) * D + dt * 16 + m] = acc[i];
    }
}

// ---------------------------------------------------------------------------
// Kernel 2: attention coefficients  es = h . a[:,0],  ed = h . a[:,1]
// ---------------------------------------------------------------------------
template<int D>
__global__ void attn_coef_kernel(const float* __restrict__ Hh,
                                 const float* __restrict__ a,
                                 float* __restrict__ es, float* __restrict__ ed,
                                 int total) {
    int idx = blockIdx.x * blockDim.x + threadIdx.x;
    if (idx >= total) return;
    int head = (idx / NN) % HH;
    const float4* hrow = (const float4*)(Hh + (size_t)idx * D);
    const float4* a0 = (const float4*)(a + (size_t)head * 2 * D);
    const float4* a1 = a0 + D / 4;
    float s0 = 0.f, s1 = 0.f;
#pragma unroll
    for (int d = 0; d < D / 4; ++d) {
        float4 v = hrow[d], p = a0[d], q = a1[d];
        s0 += v.x * p.x + v.y * p.y + v.z * p.z + v.w * p.w;
        s1 += v.x * q.x + v.y * q.y + v.z * q.z + v.w * q.w;
    }
    es[idx] = s0;
    ed[idx] = s1;
}

// ---------------------------------------------------------------------------
// Kernel 3: masked-softmax attention + aggregation (two pass, async double
// buffered).  One wave per (16-row block, head, t).  o = softmax(e) @ h with
// e = leaky_relu(es_i + ed_j) masked by A>0.  P built directly in WMMA
// A-fragment layout; P@h accumulated with v_wmma_f32_16x16x4_f32.
// ---------------------------------------------------------------------------
template<int D, bool CONCAT>
__global__ __launch_bounds__(32)
void gat_attn_kernel(const float* __restrict__ A, const float* __restrict__ Hh,
                     const float* __restrict__ es, const float* __restrict__ ed,
                     float* __restrict__ Z) {
    constexpr int NB = NN / 16;                 // 125 column chunks
    constexpr int KA = 2;                       // async b128 per 16x16 A tile
    constexpr int KH = (16 * D * 4) / 512;      // async b128 per 16xD h tile

    const int rb = blockIdx.x, head = blockIdx.y, t = blockIdx.z;
    const int lane = threadIdx.x;
    const int m  = lane & 15;
    const int hi = lane >> 4;

    __shared__ float As[2][256];
    __shared__ float hs[2][16 * D];

    const float* At  = A  + (size_t)t * NN * NN;
    const float* ht  = Hh + (((size_t)t * HH + head) * NN) * D;
    const float* est = es + ((size_t)t * HH + head) * NN;
    const float* edt = ed + ((size_t)t * HH + head) * NN;

    // per-lane async sources: A tile rows are strided, h tile is contiguous
    const float* Asrc0 = At + (size_t)(rb * 16 + (lane >> 2)) * NN + (lane & 3) * 4;
    const float* Asrc1 = Asrc0 + (size_t)8 * NN;
    const float* Hsrc  = ht + lane * 4;

    const unsigned AsBase = (unsigned)(uintptr_t)&As[0][0];
    const unsigned HsBase = (unsigned)(uintptr_t)&hs[0][0];

    auto issueA = [&](int cb, int buf) {
        unsigned l = AsBase + buf * 1024 + lane * 16;
        async_b128(l,       Asrc0 + cb * 16);
        async_b128(l + 512, Asrc1 + cb * 16);
    };
    auto issueH = [&](int cb, int buf) {
        unsigned l = HsBase + buf * (16 * D * 4) + lane * 16;
        const float* g = Hsrc + (size_t)cb * 16 * D;
#pragma unroll
        for (int q = 0; q < KH; ++q)
            async_b128(l + q * 512, g + q * 128);
    };

    const int row = rb * 16 + m;
    const float my_es = est[row];

    // ---- pass 1: per-row max & sum over unmasked columns -------------------
    float mx = -3.0e38f, sm = 0.0f;
    issueA(0, 0);
    for (int cb = 0; cb < NB; ++cb) {
        const int buf = cb & 1;
        if (cb + 1 < NB) { issueA(cb + 1, buf ^ 1); wait_async<KA>(); }
        else             { wait_async<0>(); }
        const float4* Arow = (const float4*)&As[buf][m * 16 + hi * 8];
        float4 a0 = Arow[0], a1 = Arow[1];
        const float4* edq = (const float4*)(edt + cb * 16 + hi * 8);
        float4 e0q = edq[0], e1q = edq[1];
        float av[8] = {a0.x, a0.y, a0.z, a0.w, a1.x, a1.y, a1.z, a1.w};
        float ev[8] = {e0q.x, e0q.y, e0q.z, e0q.w, e1q.x, e1q.y, e1q.z, e1q.w};
#pragma unroll
        for (int j = 0; j < 8; ++j) {
            if (av[j] > 0.0f) {
                float e = my_es + ev[j];
                e = (e > 0.0f) ? e : ALPHA_L * e;
                if (e > mx) { sm = sm * __expf(mx - e) + 1.0f; mx = e; }
                else        { sm += __expf(e - mx); }
            }
        }
    }
    // combine the two half-lane partials: lane L owns row m, pairs with L^16
    {
        float mx2 = __shfl_xor(mx, 16, 32);
        float sm2 = __shfl_xor(sm, 16, 32);
        float M = fmaxf(mx, mx2);
        sm = sm * __expf(mx - M) + sm2 * __expf(mx2 - M);
        mx = M;
    }
    const float Mrow = mx, Sinv = 1.0f / sm;

    // ---- pass 2: P @ h accumulated with WMMA, DMA double buffered ----------
    v8f acc[D / 16] = {};
    issueA(0, 0);
    issueH(0, 0);
    for (int cb = 0; cb < NB; ++cb) {
        const int buf = cb & 1;
        if (cb + 1 < NB) {
            issueA(cb + 1, buf ^ 1);
            issueH(cb + 1, buf ^ 1);
            wait_async<KA + KH>();
        } else {
            wait_async<0>();
        }
        const float2* Arow2 = (const float2*)&As[buf][m * 16];
        const float4* edq   = (const float4*)(edt + cb * 16);
#pragma unroll
        for (int kb = 0; kb < 4; ++kb) {
            const int k0 = kb * 4 + hi * 2;
            float2 ap  = Arow2[2 * kb + hi];       // A[m][k0], A[m][k0+1]
            float4 edk = edq[kb];                  // ed for cols 4kb..4kb+3
            float ea = hi ? edk.z : edk.x;
            float eb = hi ? edk.w : edk.y;
            float e0 = my_es + ea; e0 = (e0 > 0.f) ? e0 : ALPHA_L * e0;
            float e1 = my_es + eb; e1 = (e1 > 0.f) ? e1 : ALPHA_L * e1;
            v2f a;
            a.x = (ap.x > 0.f) ? __expf(e0 - Mrow) * Sinv : 0.0f;
            a.y = (ap.y > 0.f) ? __expf(e1 - Mrow) * Sinv : 0.0f;
#pragma unroll
            for (int dt = 0; dt < D / 16; ++dt) {
                v2f b;
                b.x = hs[buf][k0 * D + dt * 16 + m];
                b.y = hs[buf][(k0 + 1) * D + dt * 16 + m];
                acc[dt] = __builtin_amdgcn_wmma_f32_16x16x4_f32(
                    false, a, false, b, (short)0, acc[dt], false, false);
            }
        }
    }

    // ---- write out ---------------------------------------------------------
    if (CONCAT) {
        float* zt = Z + ((size_t)t * NN + (size_t)rb * 16) * (HH * D) + head * D;
#pragma unroll
        for (int dt = 0; dt < D / 16; ++dt)
#pragma unroll
            for (int i = 0; i < 8; ++i) {
                float v = acc[dt][i];
                v = (v > 0.0f) ? v : (__expf(v) - 1.0f);   // ELU
                zt[(size_t)(i + hi * 8) * (HH * D) + dt * 16 + m] = v;
            }
    } else {
        float* zt = Z + ((size_t)t * NN + (size_t)rb * 16) * D;
#pragma unroll
        for (int dt = 0; dt < D / 16; ++dt)
#pragma unroll
            for (int i = 0; i < 8; ++i)
                atomicAdd(&zt[(size_t)(i + hi * 8) * D + dt * 16 + m],
                          acc[dt][i] * (1.0f / HH));
    }
}

// ---------------------------------------------------------------------------
// Kernel 4: per-node GRU step (memory-bound per-node DxD matvecs).
//   Wg/Ug: [3,N,D,D], bg: [3,N,D], Zt: [N,D], Qin/Qout: [N,D]
// ---------------------------------------------------------------------------
template<int D>
__global__ __launch_bounds__(D)
void gru_step_kernel(const float* __restrict__ Wg, const float* __restrict__ Ug,
                     const float* __restrict__ bg, const float* __restrict__ Zt,
                     const float* __restrict__ Qin, float* __restrict__ Qout) {
    const int n = blockIdx.x;
    const int i = threadIdx.x;
    __shared__ float zs[D], qs[D], rqs[D];

    const size_t nd = (size_t)n * D + i;
    zs[i] = Zt[nd];
    qs[i] = Qin[nd];
    __syncthreads();

    const size_t gs4 = (size_t)NN * D * D / 4;      // gate stride in float4
    const float4* Wn = (const float4*)(Wg + (size_t)n * D * D + (size_t)i * D);
    const float4* Un = (const float4*)(Ug + (size_t)n * D * D + (size_t)i * D);
    const float4* zs4 = (const float4*)zs;
    const float4* qs4 = (const float4*)qs;

    float wz0 = bg[(size_t)0 * NN * D + nd];
    float wz1 = bg[(size_t)1 * NN * D + nd];
    float wz2 = bg[(size_t)2 * NN * D + nd];
    float uq0 = 0.f, uq1 = 0.f;
#pragma unroll 2
    for (int j = 0; j < D / 4; ++j) {
        float4 z = zs4[j], q = qs4[j];
        float4 w0 = Wn[j], w1 = Wn[gs4 + j], w2 = Wn[2 * gs4 + j];
        float4 u0 = Un[j], u1 = Un[gs4 + j];
        wz0 += w0.x * z.x + w0.y * z.y + w0.z * z.z + w0.w * z.w;
        wz1 += w1.x * z.x + w1.y * z.y + w1.z * z.z + w1.w * z.w;
        wz2 += w2.x * z.x + w2.y * z.y + w2.z * z.z + w2.w * z.w;
        uq0 += u0.x * q.x + u0.y * q.y + u0.z * q.z + u0.w * q.w;
        uq1 += u1.x * q.x + u1.y * q.y + u1.z * q.z + u1.w * q.w;
    }
    float upd = 1.0f / (1.0f + __expf(-(wz0 + uq0)));
    float rst = 1.0f / (1.0f + __expf(-(wz1 + uq1)));
    rqs[i] = rst * qs[i];
    __syncthreads();

    const float4* rqs4 = (const float4*)rqs;
    float uq2 = 0.f;
#pragma unroll 2
    for (int j = 0; j < D / 4; ++j) {
        float4 u2 = Un[2 * gs4 + j];
        float4 r = rqs4[j];
        uq2 += u2.x * r.x + u2.y * r.y + u2.z * r.z + u2.w * r.w;
    }
    float hcap = tanhf(wz2 + uq2);
    Qout[nd] = (1.0f - upd) * qs[i] + upd * hcap;
}

// ---------------------------------------------------------------------------
// Host launcher
// ---------------------------------------------------------------------------
extern "C" void kernel_launch(void* const* d_in, const int* in_sizes, int n_in,
                              void* d_out, int out_size, void* d_ws, size_t ws_size,
                              hipStream_t stream) {
    const float* A      = (const float*)d_in[0];   // [T,N,N]
    const float* feats  = (const float*)d_in[1];   // [T,N,32]
    const float* gat1_W = (const float*)d_in[2];   // [4,32,16]
    const float* gat1_a = (const float*)d_in[3];   // [4,2,16]
    const float* gru1_W = (const float*)d_in[4];   // [3,N,64,64]
    const float* gru1_U = (const float*)d_in[5];
    const float* gru1_b = (const float*)d_in[6];   // [3,N,64]
    const float* q1     = (const float*)d_in[7];   // [N,64]
    const float* gat2_W = (const float*)d_in[8];   // [4,64,32]
    const float* gat2_a = (const float*)d_in[9];   // [4,2,32]
    const float* gru2_W = (const float*)d_in[10];  // [3,N,32,32]
    const float* gru2_U = (const float*)d_in[11];
    const float* gru2_b = (const float*)d_in[12];
    const float* q2     = (const float*)d_in[13];  // [N,32]
    float* out = (float*)d_out;                    // [T*N,32]

    // Workspace layout (floats)
    float* ws  = (float*)d_ws;
    float* h1  = ws;                                 // T*H*N*16
    float* es1 = h1  + (size_t)TT * HH * NN * 16;
    float* ed1 = es1 + (size_t)TT * HH * NN;
    float* Z1  = ed1 + (size_t)TT * HH * NN;         // T*N*64
    float* s1  = Z1  + (size_t)TT * NN * 64;         // T*N*64
    float* h2  = s1  + (size_t)TT * NN * 64;         // T*H*N*32
    float* es2 = h2  + (size_t)TT * HH * NN * 32;
    float* ed2 = es2 + (size_t)TT * HH * NN;
    float* Z2  = ed2 + (size_t)TT * HH * NN;         // T*N*32

    const dim3 gatGrid(NN / 16, HH, TT);

    // ------------------ Layer 1 (concat, D_head=16, F=32) -------------------
    hproj_kernel<32, 16><<<gatGrid, 32, 0, stream>>>(feats, gat1_W, h1);
    {
        int total = TT * HH * NN;
        attn_coef_kernel<16><<<(total + 255) / 256, 256, 0, stream>>>(
            h1, gat1_a, es1, ed1, total);
    }
    gat_attn_kernel<16, true><<<gatGrid, 32, 0, stream>>>(A, h1, es1, ed1, Z1);
    for (int t = 0; t < TT; ++t) {
        const float* Qin = (t == 0) ? q1 : (s1 + (size_t)(t - 1) * NN * 64);
        gru_step_kernel<64><<<NN, 64, 0, stream>>>(
            gru1_W, gru1_U, gru1_b, Z1 + (size_t)t * NN * 64, Qin,
            s1 + (size_t)t * NN * 64);
    }

    // ------------------ Layer 2 (head-mean, D_head=32, F=64) ----------------
    hproj_kernel<64, 32><<<gatGrid, 32, 0, stream>>>(s1, gat2_W, h2);
    {
        int total = TT * HH * NN;
        attn_coef_kernel<32><<<(total + 255) / 256, 256, 0, stream>>>(
            h2, gat2_a, es2, ed2, total);
    }
    hipMemsetAsync(Z2, 0, (size_t)TT * NN * 32 * sizeof(float), stream);
    gat_attn_kernel<32, false><<<gatGrid, 32, 0, stream>>>(A, h2, es2, ed2, Z2);
    for (int t = 0; t < TT; ++t) {
        const float* Qin = (t == 0) ? q2 : (out + (size_t)(t - 1) * NN * 32);
        gru_step_kernel<32><<<NN, 32, 0, stream>>>(
            gru2_W, gru2_U, gru2_b, Z2 + (size_t)t * NN * 32, Qin,
            out + (size_t)t * NN * 32);
    }
}